// LI_82927228552103
// MI455X (gfx1250) — compile-verified
//
#include <hip/hip_runtime.h>

// Leaky-integrator scan  mem[t] = DECAY*mem[t-1] + inputs[t-1]
// reformulated as chunked lower-triangular matmul on fp32 WMMA.
//   T=1024, B=64, F=1024  ->  CH = B*F = 65536 independent channels.
// Memory-bound: 512MB streaming traffic -> ~22us floor @ 23.3 TB/s.
// NT cache hints: data is touched exactly once, bigger than the 192MB L2.

typedef __attribute__((ext_vector_type(2))) float v2f;
typedef __attribute__((ext_vector_type(8))) float v8f;

#define LI_DECAY   0.95f          // 1 - 1/20
#define LI_T       1024
#define LI_CH      65536          // B*F
#define LI_CHUNK   16
#define LI_NCHUNK  (LI_T / LI_CHUNK)
#define WAVES_PER_BLOCK 4

// One 16-timestep chunk: acc = decay^(row+1)*carry ; acc += L * Xchunk ;
// store 16x16 tile ; carry = row 15.  FIRST peels the t==0 shift edge so the
// steady-state loop body has no compare/select.
template <bool FIRST>
__device__ __forceinline__ void li_chunk(int c, const float* __restrict__ x,
                                         float* __restrict__ out, long chOfs,
                                         int col, int hi, const v2f* A,
                                         const float* fac, float& carry) {
  // B = X[4g:4g+4, :] in WMMA fp32 B layout:
  //   VGPR v, lanes 0-15: row K = 4g + v      (N = lane)
  //   VGPR v, lanes16-31: row K = 4g + 2 + v  (N = lane-16)
  // Row j is time t = c*16 + j, whose value is inputs[t-1] (0 for t==0).
  v2f Bm[4];
  for (int g = 0; g < 4; ++g) {
    const int t0 = c * LI_CHUNK + 4 * g + 2 * hi;  // time of B element .x
    float b0;
    if (FIRST) {
      const int tm0 = (t0 > 0) ? (t0 - 1) : 0;     // clamped input index
      b0 = __builtin_nontemporal_load(&x[(long)tm0 * LI_CH + chOfs]);
      if (t0 == 0) b0 = 0.0f;                      // select, EXEC stays all-1s
    } else {
      b0 = __builtin_nontemporal_load(&x[(long)(t0 - 1) * LI_CH + chOfs]);
    }
    const float b1 =                               // time t0+1 -> input t0
        __builtin_nontemporal_load(&x[(long)t0 * LI_CH + chOfs]);
    Bm[g].x = b0;
    Bm[g].y = b1;
  }

  v8f acc;
  for (int r = 0; r < 8; ++r) acc[r] = fac[r] * carry;

  for (int g = 0; g < 4; ++g) {
    acc = __builtin_amdgcn_wmma_f32_16x16x4_f32(
        /*neg_a=*/false, A[g], /*neg_b=*/false, Bm[g],
        /*c_mod=*/(short)0, acc, /*reuse_a=*/false, /*reuse_b=*/false);
  }

  // Store the 16x16 tile: VGPR r holds row (r + 8*hi), col = lane%16.
  for (int r = 0; r < 8; ++r) {
    const long row = c * LI_CHUNK + r + 8 * hi;
    __builtin_nontemporal_store(acc[r], &out[row * LI_CH + chOfs]);
  }

  // New carry = row 15 = acc[7] of lanes 16-31; broadcast per-column.
  carry = __shfl(acc[7], 16 + col, 32);
}

__global__ __launch_bounds__(32 * WAVES_PER_BLOCK, 1)
void li_scan_wmma_kernel(const float* __restrict__ x, float* __restrict__ out) {
  const int lane  = threadIdx.x & 31;
  const int wave  = threadIdx.x >> 5;
  const int strip = blockIdx.x * WAVES_PER_BLOCK + wave;  // 16-channel strip
  const int col   = lane & 15;   // column N within the 16x16 tile
  const int hi    = lane >> 4;   // 0: lanes 0-15, 1: lanes 16-31
  const long chOfs = (long)strip * 16 + col;              // this lane's channel
  const int M     = col;         // A-matrix row owned by this lane

  // A = L[:, 4g:4g+4] in WMMA fp32 A layout:
  //   VGPR v, lane: element (M = lane%16, K = 4g + v + 2*hi)
  //   L[m][k] = DECAY^(m-k) for k<=m else 0
  v2f A[4];
  for (int g = 0; g < 4; ++g) {
    const int k0 = 4 * g + 2 * hi;
    A[g].x = (k0     <= M) ? __powf(LI_DECAY, (float)(M - k0))     : 0.0f;
    A[g].y = (k0 + 1 <= M) ? __powf(LI_DECAY, (float)(M - k0 - 1)) : 0.0f;
  }

  // Carry-in scale per accumulator VGPR r: DECAY^(row+1), row = r + 8*hi.
  float fac[8];
  for (int r = 0; r < 8; ++r)
    fac[r] = __powf(LI_DECAY, (float)(r + 8 * hi + 1));

  float carry = 0.0f;  // mem value of the last row of the previous chunk

  li_chunk<true>(0, x, out, chOfs, col, hi, A, fac, carry);
  for (int c = 1; c < LI_NCHUNK; ++c)
    li_chunk<false>(c, x, out, chOfs, col, hi, A, fac, carry);
}

extern "C" void kernel_launch(void* const* d_in, const int* in_sizes, int n_in,
                              void* d_out, int out_size, void* d_ws, size_t ws_size,
                              hipStream_t stream) {
  (void)in_sizes; (void)n_in; (void)d_ws; (void)ws_size; (void)out_size;
  const float* x = (const float*)d_in[0];
  float* out = (float*)d_out;
  const int nStrips = LI_CH / 16;                       // 4096 waves
  const int nBlocks = nStrips / WAVES_PER_BLOCK;        // 1024 blocks
  li_scan_wmma_kernel<<<nBlocks, 32 * WAVES_PER_BLOCK, 0, stream>>>(x, out);
}